// LNOBlock_73830487818861
// MI455X (gfx1250) — compile-verified
//
#include <hip/hip_runtime.h>
#include <math.h>

typedef __attribute__((ext_vector_type(2))) float v2f;
typedef __attribute__((ext_vector_type(4))) float v4f;
typedef __attribute__((ext_vector_type(8))) float v8f;

#define CDIM   128
#define RNK    64
#define NPTS   8192
#define BATCH  2
#define CHUNKS 32                 /* chunks per batch for the Gram kernel */
#define CROWS  (NPTS / CHUNKS)    /* 256 rows per chunk */
#define TROWS  64                 /* rows staged in LDS per barrier in k_gram */

__device__ __forceinline__ float gelu_exact(float v) {
    return 0.5f * v * (1.0f + erff(v * 0.7071067811865475f));
}

// ---------------------------------------------------------------------------
// Kernel 1: per-chunk partial Gram G_part = X_chunk^T X_chunk (f32 WMMA 16x16x4)
// LDS holds x rows pair-interleaved: xsp[p*256 + c*2 + {0,1}] = x[2p / 2p+1, c]
// so every A/B fragment is one aligned ds_load_b64 (no repack movs).
// grid = BATCH*CHUNKS blocks of 256 threads (8 waves).
// ---------------------------------------------------------------------------
__global__ __launch_bounds__(256) void k_gram(const float* __restrict__ x,
                                              float* __restrict__ Gpart,
                                              float* __restrict__ Spart) {
    __shared__ float xsp[(TROWS / 2) * 256];        // 32 KB
    const int tid  = threadIdx.x;
    const int wave = tid >> 5;
    const int lane = tid & 31;
    const int half = lane >> 4;                     // 0: K pair 2ks, 1: pair 2ks+1
    const int l16  = lane & 15;
    const int bid  = blockIdx.x;
    const int b    = bid >> 5;                      // CHUNKS == 32
    const int n0   = (bid & 31) * CROWS;
    const float* xb = x + (size_t)b * NPTS * CDIM;

    v8f acc[8];
#pragma unroll
    for (int j = 0; j < 8; ++j) acc[j] = (v8f){0.f,0.f,0.f,0.f,0.f,0.f,0.f,0.f};
    float cacc[4] = {0.f, 0.f, 0.f, 0.f};           // col partials, cols 4*(tid&31)+q
    const int ci0 = wave * 16;

    for (int it = 0; it < CROWS / TROWS; ++it) {
        const int rbase = n0 + it * TROWS;
        // stage TROWS x 128 block: b128 global loads, pair-interleaved b64 LDS stores
        for (int g = tid; g < (TROWS / 2) * 32; g += 256) {
            const int p  = g >> 5;                  // pair-row
            const int c4 = g & 31;                  // fixed per thread (==tid&31)
            const v4f fa = *(const v4f*)&xb[(size_t)(rbase + 2 * p) * CDIM + 4 * c4];
            const v4f fb = *(const v4f*)&xb[(size_t)(rbase + 2 * p + 1) * CDIM + 4 * c4];
#pragma unroll
            for (int q = 0; q < 4; ++q) {
                *(v2f*)&xsp[p * 256 + (4 * c4 + q) * 2] = (v2f){fa[q], fb[q]};
                cacc[q] += fa[q] + fb[q];
            }
        }
        __syncthreads();
        for (int ks = 0; ks < TROWS / 4; ++ks) {    // K advances 4 rows per step
            const int prow = 2 * ks + half;
            const v2f a = *(const v2f*)&xsp[prow * 256 + (ci0 + l16) * 2];
#pragma unroll
            for (int j = 0; j < 8; ++j) {
                const v2f bb = *(const v2f*)&xsp[prow * 256 + (16 * j + l16) * 2];
                acc[j] = __builtin_amdgcn_wmma_f32_16x16x4_f32(
                    false, a, false, bb, (short)0, acc[j], false, false);
            }
        }
        __syncthreads();
    }
    // D layout: VGPR i, lanes 0-15 -> (M=i, N=lane); lanes 16-31 -> (M=8+i, N=lane-16)
    float* Gp = Gpart + (size_t)bid * (CDIM * CDIM);
#pragma unroll
    for (int j = 0; j < 8; ++j) {
#pragma unroll
        for (int i = 0; i < 8; ++i) {
            const int row = ci0 + half * 8 + i;
            const int col = j * 16 + l16;
            Gp[row * CDIM + col] = acc[j][i];
        }
    }
#pragma unroll
    for (int q = 0; q < 4; ++q) Spart[bid * 1024 + tid * 4 + q] = cacc[q];
}

// ---------------------------------------------------------------------------
// Kernel 2: deterministic fixed-order reduction of chunk partials -> G[b], S[b].
// grid = BATCH*128 blocks of 32 threads (b128 loads).
// ---------------------------------------------------------------------------
__global__ __launch_bounds__(32) void k_reduce(const float* __restrict__ Gpart,
                                               const float* __restrict__ Spart,
                                               float* __restrict__ G,
                                               float* __restrict__ S) {
    const int b   = blockIdx.x >> 7;
    const int row = blockIdx.x & 127;
    const int c4  = threadIdx.x;                    // 0..31
    v4f acc = (v4f){0.f, 0.f, 0.f, 0.f};
    for (int ch = 0; ch < CHUNKS; ++ch)
        acc += *(const v4f*)&Gpart[(size_t)(b * CHUNKS + ch) * (CDIM * CDIM)
                                   + row * CDIM + 4 * c4];
    *(v4f*)&G[b * CDIM * CDIM + row * CDIM + 4 * c4] = acc;
    if (row == 0) {
#pragma unroll
        for (int q = 0; q < 4; ++q) {
            float s = 0.f;
            for (int ch = 0; ch < CHUNKS; ++ch)
                for (int w = 0; w < 8; ++w)
                    s += Spart[(b * CHUNKS + ch) * 1024 + (w * 32 + c4) * 4 + q];
            S[b * CDIM + 4 * c4 + q] = s;
        }
    }
}

// ---------------------------------------------------------------------------
// Kernel 3: I[b,r,c] = (1/N)*sum_k psi_w[k,r,c]*G[b,k,c] + psi_b[r,c]*S[b,c]/N
// grid = BATCH*RNK blocks of 128 threads; thread = (kq, c4); b128 streams of
// psi_w (read exactly once, 4 MB); deterministic 4-way LDS combine over kq.
// ---------------------------------------------------------------------------
__global__ __launch_bounds__(128) void k_integral(const float* __restrict__ psi_w,
                                                  const float* __restrict__ psi_b,
                                                  const float* __restrict__ G,
                                                  const float* __restrict__ S,
                                                  float* __restrict__ I) {
    __shared__ v4f part[4][32];
    const int b  = blockIdx.x >> 6;                 // RNK == 64
    const int r  = blockIdx.x & 63;
    const int kq = threadIdx.x >> 5;
    const int c4 = threadIdx.x & 31;
    const float* wbase = psi_w + r * CDIM + 4 * c4;
    const float* gbase = G + b * CDIM * CDIM + 4 * c4;
    v4f acc = (v4f){0.f, 0.f, 0.f, 0.f};
    for (int k = kq * 32; k < kq * 32 + 32; ++k)
        acc += (*(const v4f*)&wbase[(size_t)k * (RNK * CDIM)]) *
               (*(const v4f*)&gbase[k * CDIM]);
    part[kq][c4] = acc;
    __syncthreads();
    if (kq == 0) {
        const float invN = 1.0f / (float)NPTS;
        v4f tot = ((part[0][c4] + part[1][c4]) + part[2][c4]) + part[3][c4];
        const v4f pb = *(const v4f*)&psi_b[r * CDIM + 4 * c4];
        const v4f sv = *(const v4f*)&S[b * CDIM + 4 * c4];
        *(v4f*)&I[b * RNK * CDIM + r * CDIM + 4 * c4] = tot * invN + pb * sv * invN;
    }
}

// ---------------------------------------------------------------------------
// Kernel 4: Weff[b] = W_w + phi_w @ I[b], beff[b] = W_b + phi_b @ I[b]. Tiny.
// grid = BATCH*129 blocks of 128 threads (row 128 == bias row).
// ---------------------------------------------------------------------------
__global__ __launch_bounds__(128) void k_weff(const float* __restrict__ W_w,
                                              const float* __restrict__ W_b,
                                              const float* __restrict__ phi_w,
                                              const float* __restrict__ phi_b,
                                              const float* __restrict__ I,
                                              float* __restrict__ Weff,
                                              float* __restrict__ beff) {
    const int b   = blockIdx.x / 129;
    const int row = blockIdx.x % 129;
    const int c   = threadIdx.x;
    const float* Ib = I + b * RNK * CDIM;
    if (row < CDIM) {
        float acc = W_w[row * CDIM + c];
        for (int rr = 0; rr < RNK; ++rr)
            acc += phi_w[row * RNK + rr] * Ib[rr * CDIM + c];
        Weff[b * CDIM * CDIM + row * CDIM + c] = acc;
    } else {
        float acc = W_b[c];
        for (int rr = 0; rr < RNK; ++rr)
            acc += phi_b[rr] * Ib[rr * CDIM + c];
        beff[b * CDIM + c] = acc;
    }
}

// ---------------------------------------------------------------------------
// Kernel 5: out = gelu(x @ Weff[b] + beff[b]) — f32 WMMA GEMM, fused epilogue.
// ws: Weff pair-interleaved (b64 B-fragments); xs: x rows, stride 132 dwords
// (even & mult-of-4 -> aligned b64 A-fragments and aligned b128 reuse).
// Epilogue: gelu -> own-wave LDS rows -> coalesced b128 global stores.
// grid = 128 WGs of 256 threads (8 waves x 16 rows).
// ---------------------------------------------------------------------------
#define XSTR 132
__global__ __launch_bounds__(256) void k_out(const float* __restrict__ x,
                                             const float* __restrict__ Weff,
                                             const float* __restrict__ beff,
                                             float* __restrict__ out) {
    extern __shared__ float smem[];
    float* ws = smem;                        // 16384 dwords, pair-interleaved Weff
    float* xs = smem + CDIM * CDIM;          // 128*132 dwords
    const int tid  = threadIdx.x;
    const int wave = tid >> 5;
    const int lane = tid & 31;
    const int half = lane >> 4;
    const int l16  = lane & 15;
    const int b    = blockIdx.x >> 6;        // 64 WGs per batch
    const size_t row0 = (size_t)blockIdx.x * 128;
    const float* Wb = Weff + b * CDIM * CDIM;

    // stage Weff pair-interleaved: ws[p*256 + c*2 + {0,1}] = W[2p / 2p+1, c]
    for (int g = tid; g < 64 * 32; g += 256) {
        const int p  = g >> 5;
        const int c4 = g & 31;
        const v4f fa = *(const v4f*)&Wb[(2 * p) * CDIM + 4 * c4];
        const v4f fb = *(const v4f*)&Wb[(2 * p + 1) * CDIM + 4 * c4];
#pragma unroll
        for (int q = 0; q < 4; ++q)
            *(v2f*)&ws[p * 256 + (4 * c4 + q) * 2] = (v2f){fa[q], fb[q]};
    }
    // stage x block row-major (A fragment = adjacent columns -> natural b64)
    for (int i4 = tid; i4 < (CDIM * CDIM) / 4; i4 += 256) {
        const int r  = i4 >> 5;
        const int c4 = i4 & 31;
        *(v4f*)&xs[r * XSTR + 4 * c4] = *(const v4f*)&x[(row0 + r) * CDIM + 4 * c4];
    }
    __syncthreads();

    v8f acc[8];
#pragma unroll
    for (int j = 0; j < 8; ++j) acc[j] = (v8f){0.f,0.f,0.f,0.f,0.f,0.f,0.f,0.f};
    const int r0 = wave * 16;                // this wave's 16 output rows

    for (int ks = 0; ks < CDIM / 4; ++ks) {
        // A[m,k] = x[r0+m, 4ks+k]: lanes 0-15 hold K={0,1}, 16-31 hold K={2,3}
        const v2f a = *(const v2f*)&xs[(r0 + l16) * XSTR + 4 * ks + 2 * half];
#pragma unroll
        for (int j = 0; j < 8; ++j) {
            // B[k,n] = Weff[4ks+k, 16j+n] from pair-interleaved ws
            const v2f bb = *(const v2f*)&ws[(2 * ks + half) * 256 + (16 * j + l16) * 2];
            acc[j] = __builtin_amdgcn_wmma_f32_16x16x4_f32(
                false, a, false, bb, (short)0, acc[j], false, false);
        }
    }

    // epilogue: bias + exact gelu into own-wave LDS rows, then b128 stores
    const float* bv = beff + b * CDIM;
#pragma unroll
    for (int j = 0; j < 8; ++j) {
        const int col  = j * 16 + l16;
        const float bias = bv[col];
#pragma unroll
        for (int i = 0; i < 8; ++i) {
            const int row = r0 + half * 8 + i;
            xs[row * XSTR + col] = gelu_exact(acc[j][i] + bias);
        }
    }
    // per-wave, in-order LDS: no barrier needed (only own rows touched)
#pragma unroll
    for (int v = 0; v < 16; ++v) {
        const v4f val = *(const v4f*)&xs[(r0 + v) * XSTR + 4 * lane];
        *(v4f*)&out[(row0 + r0 + v) * CDIM + 4 * lane] = val;
    }
}

// ---------------------------------------------------------------------------
extern "C" void kernel_launch(void* const* d_in, const int* in_sizes, int n_in,
                              void* d_out, int out_size, void* d_ws, size_t ws_size,
                              hipStream_t stream) {
    (void)in_sizes; (void)n_in; (void)out_size; (void)ws_size;
    const float* x     = (const float*)d_in[0];
    const float* W_w   = (const float*)d_in[1];
    const float* W_b   = (const float*)d_in[2];
    const float* phi_w = (const float*)d_in[3];
    const float* phi_b = (const float*)d_in[4];
    const float* psi_w = (const float*)d_in[5];
    const float* psi_b = (const float*)d_in[6];
    float* out = (float*)d_out;

    // workspace layout (floats)
    float* ws    = (float*)d_ws;
    float* Gpart = ws;                                   // 64*16384
    float* Spart = Gpart + BATCH * CHUNKS * CDIM * CDIM; // 64*1024
    float* G     = Spart + BATCH * CHUNKS * 1024;        // 2*16384
    float* S     = G + BATCH * CDIM * CDIM;              // 2*128
    float* I     = S + BATCH * CDIM;                     // 2*64*128
    float* Weff  = I + BATCH * RNK * CDIM;               // 2*16384
    float* beff  = Weff + BATCH * CDIM * CDIM;           // 2*128

    k_gram    <<<BATCH * CHUNKS, 256, 0, stream>>>(x, Gpart, Spart);
    k_reduce  <<<BATCH * CDIM,    32, 0, stream>>>(Gpart, Spart, G, S);
    k_integral<<<BATCH * RNK,    128, 0, stream>>>(psi_w, psi_b, G, S, I);
    k_weff    <<<BATCH * 129,    128, 0, stream>>>(W_w, W_b, phi_w, phi_b, I, Weff, beff);

    const size_t shmem = (size_t)(CDIM * CDIM + CDIM * XSTR) * sizeof(float); // 133120 B
    (void)hipFuncSetAttribute((const void*)k_out,
                              hipFuncAttributeMaxDynamicSharedMemorySize, (int)shmem);
    k_out     <<<BATCH * (NPTS / 128), 256, shmem, stream>>>(x, Weff, beff, out);
}